// InputEmbedder_36060545417651
// MI455X (gfx1250) — compile-verified
//
#include <hip/hip_runtime.h>
#include <hip/hip_bf16.h>
#include <cstdint>

typedef __attribute__((ext_vector_type(2))) float v2f;
typedef __attribute__((ext_vector_type(4))) float v4f;
typedef __attribute__((ext_vector_type(8))) float v8f;

#define R_DIM  1024
#define N_SEQ  128
#define CF     22
#define C_MSA  256
#define C_PAIR 128
#define NBINS  65

// A-operand loader: unconditional 8B load of K-pair (ka, ka+1), ka even (0..22).
// Pair valid iff ka <= 20. Clamp address, zero value by mask multiply.
// With half in {0,1} this folds to a plain b64 load for all chunks but the last.
__device__ __forceinline__ v2f load_k2_masked(const float* __restrict__ row, int ka) {
    const bool  valid = (ka + 1) < CF;
    const int   kc    = valid ? ka : 0;
    const float msk   = valid ? 1.0f : 0.0f;
    v2f v = *(const v2f*)(row + kc);
    v.x *= msk;
    v.y *= msk;
    return v;
}

// B-operand loader: address clamp only. Out-of-range K multiplies a zeroed A
// element, so the (finite) clamped weight value contributes exactly 0.
__device__ __forceinline__ v2f load_k2_clamped(const float* __restrict__ row, int ka) {
    const bool valid = (ka + 1) < CF;
    const int  kc    = valid ? ka : 0;
    return *(const v2f*)(row + kc);
}

// ---------------------------------------------------------------------------
// Kernel A: a = tf@Wa.T+ba, bv = tf@Wb.T+bb, tm2 = tf@Wm2.T+bm2
// One wave computes one 16x16 tile via V_WMMA_F32_16X16X4_F32 (6 issues, K=24).
// grid = (R/16, 8 + 8 + 16); blockIdx.y selects GEMM / column tile.
// ---------------------------------------------------------------------------
__global__ __launch_bounds__(32) void ie_precompute(
    const float* __restrict__ tf,
    const float* __restrict__ Wa,  const float* __restrict__ ba,
    const float* __restrict__ Wb,  const float* __restrict__ bb,
    const float* __restrict__ Wm2, const float* __restrict__ bm2,
    float* __restrict__ a_out, float* __restrict__ bv_out, float* __restrict__ tm2_out)
{
    const int mt = blockIdx.x;
    const int t  = blockIdx.y;
    const float* W; const float* bias; float* out; int n0, Cout;
    if (t < 8)       { W = Wa;  bias = ba;  out = a_out;   n0 = t * 16;        Cout = C_PAIR; }
    else if (t < 16) { W = Wb;  bias = bb;  out = bv_out;  n0 = (t - 8) * 16;  Cout = C_PAIR; }
    else             { W = Wm2; bias = bm2; out = tm2_out; n0 = (t - 16) * 16; Cout = C_MSA;  }

    const int lane = threadIdx.x & 31;     // bound range so pad-mask folds
    const int half = lane >> 4;            // in {0,1}: lanes 16-31 hold K-pair {2,3}
    const int l16  = lane & 15;
    const int m0   = mt * 16;

    const float* arow = tf + (m0 + l16) * CF;   // A: M = l16
    const float* brow = W  + (n0 + l16) * CF;   // B[k][n] = W[n][k]: N = l16

    v8f acc = {0.f, 0.f, 0.f, 0.f, 0.f, 0.f, 0.f, 0.f};
    #pragma unroll
    for (int k0 = 0; k0 < CF; k0 += 4) {
        const int ka = k0 + 2 * half;
        const v2f A  = load_k2_masked(arow, ka);
        const v2f Bv = load_k2_clamped(brow, ka);
        acc = __builtin_amdgcn_wmma_f32_16x16x4_f32(false, A, false, Bv,
                                                    (short)0, acc, false, false);
    }
    // C/D: VGPR r -> row m0+r (lanes 0-15) / m0+8+r (lanes 16-31), col n0+l16
    const int   col   = n0 + l16;
    const float bcol  = bias[col];
    const int   rbase = m0 + half * 8;
    #pragma unroll
    for (int r = 0; r < 8; ++r)
        out[(rbase + r) * Cout + col] = acc[r] + bcol;
}

// ---------------------------------------------------------------------------
// Kernel B: z[i,j,c] = a[j,c] + bv[i,c] + WposT[clip(ri[j]-ri[i]+32,0,64), c] + bpos[c]
// One block per row i. WposT staged in LDS (33 KB). 256 threads = 8 j-slots x
// 32 lanes x float4 over c. NT b128 stores (537 MB >> 192 MB L2).
// ---------------------------------------------------------------------------
__global__ __launch_bounds__(256) void ie_pair(
    const float* __restrict__ a, const float* __restrict__ bv,
    const int*   __restrict__ ri,
    const float* __restrict__ Wpos, const float* __restrict__ bpos,
    float* __restrict__ z)
{
    __shared__ float WT[NBINS * C_PAIR];
    const int tid = threadIdx.x;
    for (int idx = tid; idx < NBINS * C_PAIR; idx += 256) {
        const int bin = idx >> 7;
        const int cc  = idx & (C_PAIR - 1);
        WT[idx] = Wpos[cc * NBINS + bin];      // WT[bin][c] = Wpos[c][bin]
    }
    __syncthreads();

    const int i    = blockIdx.x;
    const int c4   = (tid & 31) * 4;
    const int joff = tid >> 5;                  // uniform within a wave
    const int rii  = ri[i];

    const v4f bvi  = *(const v4f*)(bv + i * C_PAIR + c4);
    const v4f bp   = *(const v4f*)(bpos + c4);
    const v4f base = bvi + bp;

    float* zrow = z + (size_t)i * R_DIM * C_PAIR;
    for (int j = joff; j < R_DIM; j += 8) {
        int d = ri[j] - rii + (NBINS / 2);
        d = d < 0 ? 0 : (d > NBINS - 1 ? NBINS - 1 : d);
        const v4f av = *(const v4f*)(a + j * C_PAIR + c4);
        const v4f wv = *(const v4f*)(&WT[d * C_PAIR + c4]);
        const v4f o  = av + base + wv;
        __builtin_nontemporal_store(o, (v4f*)(zrow + j * C_PAIR + c4));
    }
}

// ---------------------------------------------------------------------------
// Kernel C: m[n,r,c] = (msa[n,r,:] @ Wm1.T)[c] + bm1[c] + tm2[r,c]
// Flattened M = 131072 rows; one wave per 16x16 tile, 6x WMMA f32, fused
// bias + tm2 epilogue. RT stores (m = 134 MB < L2; sectors merge to lines).
// ---------------------------------------------------------------------------
__global__ __launch_bounds__(32) void ie_msa(
    const float* __restrict__ msa, const float* __restrict__ Wm1,
    const float* __restrict__ bm1, const float* __restrict__ tm2,
    float* __restrict__ m_out)
{
    const int mt   = blockIdx.x;   // 0..8191
    const int nt   = blockIdx.y;   // 0..15
    const int lane = threadIdx.x & 31;
    const int half = lane >> 4;
    const int l16  = lane & 15;
    const int m0   = mt * 16;
    const int n0   = nt * 16;

    const float* arow = msa + (size_t)(m0 + l16) * CF;
    const float* brow = Wm1 + (n0 + l16) * CF;

    v8f acc = {0.f, 0.f, 0.f, 0.f, 0.f, 0.f, 0.f, 0.f};
    #pragma unroll
    for (int k0 = 0; k0 < CF; k0 += 4) {
        const int ka = k0 + 2 * half;
        const v2f A  = load_k2_masked(arow, ka);
        const v2f Bv = load_k2_clamped(brow, ka);
        acc = __builtin_amdgcn_wmma_f32_16x16x4_f32(false, A, false, Bv,
                                                    (short)0, acc, false, false);
    }
    const int   col   = n0 + l16;
    const float bcol  = bm1[col];
    const int   rbase = m0 + half * 8;
    #pragma unroll
    for (int r = 0; r < 8; ++r) {
        const int rr   = rbase + r;                 // flat (n, r) row
        const int rmod = rr & (R_DIM - 1);          // r = rr % 1024
        m_out[(size_t)rr * C_MSA + col] = acc[r] + bcol + tm2[rmod * C_MSA + col];
    }
}

// ---------------------------------------------------------------------------
extern "C" void kernel_launch(void* const* d_in, const int* in_sizes, int n_in,
                              void* d_out, int out_size, void* d_ws, size_t ws_size,
                              hipStream_t stream) {
    const float* tf   = (const float*)d_in[0];   // (1,1024,22)
    const int*   ri   = (const int*)  d_in[1];   // (1,1024)
    const float* msa  = (const float*)d_in[2];   // (1,128,1024,22)
    const float* Wa   = (const float*)d_in[3];
    const float* ba   = (const float*)d_in[4];
    const float* Wb   = (const float*)d_in[5];
    const float* bb   = (const float*)d_in[6];
    const float* Wm1  = (const float*)d_in[7];
    const float* bm1  = (const float*)d_in[8];
    const float* Wm2  = (const float*)d_in[9];
    const float* bm2  = (const float*)d_in[10];
    const float* Wpos = (const float*)d_in[11];
    const float* bpos = (const float*)d_in[12];

    // outputs concatenated in return order: m (128*1024*256) then z (1024*1024*128)
    float* m_out = (float*)d_out;
    float* z_out = (float*)d_out + (size_t)N_SEQ * R_DIM * C_MSA;

    // workspace: a (512KB) | bv (512KB) | tm2 (1MB)
    float* a_ws   = (float*)d_ws;
    float* bv_ws  = a_ws  + R_DIM * C_PAIR;
    float* tm2_ws = bv_ws + R_DIM * C_PAIR;

    ie_precompute<<<dim3(R_DIM / 16, 32), 32, 0, stream>>>(
        tf, Wa, ba, Wb, bb, Wm2, bm2, a_ws, bv_ws, tm2_ws);

    ie_pair<<<dim3(R_DIM), 256, 0, stream>>>(
        a_ws, bv_ws, ri, Wpos, bpos, z_out);

    ie_msa<<<dim3((N_SEQ * R_DIM) / 16, C_MSA / 16), 32, 0, stream>>>(
        msa, Wm1, bm1, tm2_ws, m_out);
}